// OuterProductMean_8349416424076
// MI455X (gfx1250) — compile-verified
//
#include <hip/hip_runtime.h>

#define L_SEQ   768
#define C_IN    384
#define C_OUTER 32
#define C_OUT   128
#define EPS     1e-5f

typedef __attribute__((ext_vector_type(2))) float v2f;
typedef __attribute__((ext_vector_type(8))) float v8f;

// ---------------------------------------------------------------------------
// Phase 1: per-row LayerNorm + left/right projections (tiny: ~38 MFLOP total)
// One block per sequence position l, 128 threads (4 waves).
// ---------------------------------------------------------------------------
__global__ __launch_bounds__(128) void ln_proj_kernel(
    const float* __restrict__ act,   const float* __restrict__ mask,
    const float* __restrict__ norm_w,const float* __restrict__ norm_b,
    const float* __restrict__ wl,    const float* __restrict__ bl,
    const float* __restrict__ wr,    const float* __restrict__ br,
    float* __restrict__ left_out,    float* __restrict__ right_out)
{
    const int l = blockIdx.x;
    const int t = threadIdx.x;

    __shared__ float red[128];
    __shared__ float sn[C_IN];

    const float* arow = act + (size_t)l * C_IN;

    float s = 0.f, sq = 0.f;
    for (int c = t; c < C_IN; c += 128) { float v = arow[c]; s += v; sq += v * v; }

    red[t] = s; __syncthreads();
    for (int off = 64; off > 0; off >>= 1) { if (t < off) red[t] += red[t + off]; __syncthreads(); }
    const float mean = red[0] * (1.0f / C_IN);
    __syncthreads();

    red[t] = sq; __syncthreads();
    for (int off = 64; off > 0; off >>= 1) { if (t < off) red[t] += red[t + off]; __syncthreads(); }
    const float var  = red[0] * (1.0f / C_IN) - mean * mean;
    const float rstd = rsqrtf(var + EPS);

    for (int c = t; c < C_IN; c += 128)
        sn[c] = (arow[c] - mean) * rstd * norm_w[c] + norm_b[c];
    __syncthreads();

    if (t < 2 * C_OUTER) {
        const int  d    = t & (C_OUTER - 1);
        const bool is_r = t >= C_OUTER;
        const float* w  = is_r ? wr : wl;
        float acc = 0.f;
        for (int c = 0; c < C_IN; ++c)
            acc += sn[c] * w[c * C_OUTER + d];          // coalesced across d
        acc += is_r ? br[d] : bl[d];
        acc *= mask[l];
        (is_r ? right_out : left_out)[l * C_OUTER + d] = acc;
    }
}

// ---------------------------------------------------------------------------
// Phase 2: pair-feature GEMM via V_WMMA_F32_16X16X4_F32.
//   out[i,j,e] = sum_k pair[i,j,k] * wo[k,e] + bo[e]
//   pair[.,.,k<32]  = left[j,k]   * right[i,k]
//   pair[.,.,k>=32] = left[j,k-32]- right[i,k-32]
// Block: 256 threads = 8 waves. blockIdx.y = i, blockIdx.x = 64-wide j slab.
// Wave w owns e-block [16w,16w+16); B fragments (wo) loaded once, reused
// across 4 j-tiles. A tile built in registers from LDS-staged left/right.
// Output stores are nontemporal: 302 MB streamed once, never re-read.
// ---------------------------------------------------------------------------
__global__ __launch_bounds__(256) void pair_wmma_kernel(
    const float* __restrict__ leftb, const float* __restrict__ rightb,
    const float* __restrict__ wo,    const float* __restrict__ bo,
    float* __restrict__ out)
{
    const int i      = blockIdx.y;
    const int j_base = blockIdx.x * 64;
    const int t      = threadIdx.x;
    const int lane   = t & 31;
    const int wave   = t >> 5;         // 0..7 -> e-block
    const int n      = lane & 15;      // N-col (e) / M-row (A) within tile
    const int h      = lane >> 4;      // half-wave selector (K pairing)
    const int e0     = wave * 16;

    __shared__ float s_left[64 * C_OUTER];   // 8 KB
    __shared__ float s_right[C_OUTER];

    for (int idx = t; idx < 64 * C_OUTER; idx += 256)
        s_left[idx] = leftb[(size_t)j_base * C_OUTER + idx];
    if (t < C_OUTER) s_right[t] = rightb[(size_t)i * C_OUTER + t];
    __syncthreads();

    // B fragments: wo[k, e0+n]; ISA B layout: VGPR0 = rows K=4f+2h, VGPR1 = K+1.
    v2f b[16];
#pragma unroll
    for (int f = 0; f < 16; ++f) {
        const int k0 = 4 * f + 2 * h;
        b[f].x = wo[(k0    ) * C_OUT + e0 + n];
        b[f].y = wo[(k0 + 1) * C_OUT + e0 + n];
    }

    // right[i, d] values for this lane's fixed K pattern (reused for prod+diff)
    float rv0[8], rv1[8];
#pragma unroll
    for (int g = 0; g < 8; ++g) {
        const int d0 = 4 * g + 2 * h;
        rv0[g] = s_right[d0];
        rv1[g] = s_right[d0 + 1];
    }
    const float bias = bo[e0 + n];

    for (int jb = 0; jb < 4; ++jb) {
        // A tile row m = lane&15 (ISA: lanes 0-15 and 16-31 both carry M=0..15)
        const float* lrow = &s_left[(jb * 16 + n) * C_OUTER];

        v2f a[16];
#pragma unroll
        for (int g = 0; g < 8; ++g) {
            const int d0 = 4 * g + 2 * h;
            const float l0 = lrow[d0], l1 = lrow[d0 + 1];
            a[g].x     = l0 * rv0[g];  a[g].y     = l1 * rv1[g];   // k = d      (prod)
            a[g + 8].x = l0 - rv0[g];  a[g + 8].y = l1 - rv1[g];   // k = 32 + d (diff)
        }

        v8f acc;
#pragma unroll
        for (int r = 0; r < 8; ++r) acc[r] = bias;                 // fold +bo into C

#pragma unroll
        for (int f = 0; f < 16; ++f)                               // K = 64 in 16 steps
            acc = __builtin_amdgcn_wmma_f32_16x16x4_f32(
                false, a[f], false, b[f], (short)0, acc, false, false);

        // D layout: lane half h, VGPR r -> row m = r + 8h, col e0+n
        const int j0 = j_base + jb * 16;
#pragma unroll
        for (int r = 0; r < 8; ++r) {
            const int j = j0 + r + 8 * h;
            float* dst = &out[((size_t)i * L_SEQ + j) * C_OUT + e0 + n];
            __builtin_nontemporal_store(acc[r], dst);              // TH=NT streaming store
        }
    }
}

// ---------------------------------------------------------------------------
extern "C" void kernel_launch(void* const* d_in, const int* in_sizes, int n_in,
                              void* d_out, int out_size, void* d_ws, size_t ws_size,
                              hipStream_t stream)
{
    const float* act    = (const float*)d_in[0];
    const float* mask   = (const float*)d_in[1];
    const float* norm_w = (const float*)d_in[2];
    const float* norm_b = (const float*)d_in[3];
    const float* wl     = (const float*)d_in[4];
    const float* bl     = (const float*)d_in[5];
    const float* wr     = (const float*)d_in[6];
    const float* br     = (const float*)d_in[7];
    const float* wo     = (const float*)d_in[8];
    const float* bo     = (const float*)d_in[9];
    float*       out    = (float*)d_out;

    float* leftb  = (float*)d_ws;                   // 768*32 floats
    float* rightb = leftb + L_SEQ * C_OUTER;        // 768*32 floats

    ln_proj_kernel<<<dim3(L_SEQ), dim3(128), 0, stream>>>(
        act, mask, norm_w, norm_b, wl, bl, wr, br, leftb, rightb);

    pair_wmma_kernel<<<dim3(L_SEQ / 64, L_SEQ), dim3(256), 0, stream>>>(
        leftb, rightb, wo, bo, out);
}